// CMamba_64338610094928
// MI455X (gfx1250) — compile-verified
//
#include <hip/hip_runtime.h>
#include <math.h>

// ---------------- model dims ----------------
#define BATCH      16
#define VDIM       32
#define LSEQ       2048
#define P_LEN      16
#define NPATCH     128            // (2048-16)/16 + 1
#define D_MODEL    256
#define D_INNER    512
#define D_STATE    16
#define D_CONV     4
#define DT_RANK    16
#define N_LAYER    2
#define NB         (N_LAYER*N_LAYER)
#define F_LEN      96
#define MROWS      (BATCH*NPATCH) // 2048 token rows
#define PATCH_K    (VDIM*P_LEN)   // 512
#define HEAD_N     (VDIM*F_LEN)   // 3072
#define HEAD_K     (D_MODEL*NPATCH) // 32768
#define KCHUNK     256
#define NCHUNKS    (HEAD_K/KCHUNK)   // 128
#define NTILES_HN  (HEAD_N/16)       // 192
#define HN_GROUP   4                 // N-tiles per wave in head GEMM
#define EPSF       1e-5f

typedef __attribute__((ext_vector_type(2))) float v2f;
typedef __attribute__((ext_vector_type(4))) float v4f;
typedef __attribute__((ext_vector_type(8))) float v8f;

__device__ __forceinline__ v8f wmma_f32_4(v2f a, v2f b, v8f c) {
  // V_WMMA_F32_16X16X4_F32 : D = A(16x4) * B(4x16) + C, all fp32
  return __builtin_amdgcn_wmma_f32_16x16x4_f32(false, a, false, b, (short)0, c,
                                               false, false);
}
__device__ __forceinline__ v2f vlo(v4f v) { return __builtin_shufflevector(v, v, 0, 1); }
__device__ __forceinline__ v2f vhi(v4f v) { return __builtin_shufflevector(v, v, 2, 3); }

__device__ __forceinline__ float siluf(float x) {
  return x / (1.0f + __expf(-x));
}

// ---------------------------------------------------------------------------
// Generic fp32 WMMA GEMM:  C[M,N] = A[M,K] * W[N,K]^T (+bias[n]) (+residual)
// rmode: 0 = none, 1 = R[m*ldc+n] (full residual), 2 = R[(m%128)*ldc+n] (pos)
// One wave per 16x16 tile. M,N multiples of 16; K multiple of 8.
// Per 8 K-values: one b128 A load + one b128 B load + two WMMAs, using
// virtual K-slices (lane-group k-offset baked into the pointer).
// ---------------------------------------------------------------------------
__global__ void gemm_wmma_kernel(const float* __restrict__ A, int lda,
                                 const float* __restrict__ W, int ldw,
                                 const float* __restrict__ bias,
                                 const float* __restrict__ R, int rmode,
                                 float* __restrict__ C, int ldc,
                                 int M, int N, int K) {
  const int wave = blockIdx.x * (blockDim.x >> 5) + (threadIdx.x >> 5);
  const int tiles_n = N >> 4;
  const int tiles   = (M >> 4) * tiles_n;
  if (wave >= tiles) return;                 // wave-uniform: EXEC stays full
  const int tm = wave / tiles_n, tn = wave % tiles_n;
  const int lane = threadIdx.x & 31;
  const int l15  = lane & 15;
  const int kq   = (lane >> 4) << 2;         // lane-group K offset: 0 or 4
  const int m0 = tm << 4, n0 = tn << 4;

  const float* Ap = A + (size_t)(m0 + l15) * lda + kq;
  const float* Wp = W + (size_t)(n0 + l15) * ldw + kq;  // B[k][n] = W[n][k]
  v8f acc = {};
  for (int k = 0; k < K; k += 8) {
    v4f a4 = *(const v4f*)(Ap + k);
    v4f b4 = *(const v4f*)(Wp + k);
    acc = wmma_f32_4(vlo(a4), vlo(b4), acc);   // slice {8s,8s+1,8s+4,8s+5}
    acc = wmma_f32_4(vhi(a4), vhi(b4), acc);   // slice {8s+2,8s+3,8s+6,8s+7}
  }
  const int hi = (lane >> 4) << 3;             // C layout: vgpr r -> row r/r+8
#pragma unroll
  for (int r = 0; r < 8; ++r) {
    const int m = m0 + r + hi;
    const int n = n0 + l15;
    float v = acc[r];
    if (bias)       v += bias[n];
    if (rmode == 1) v += R[(size_t)m * ldc + n];
    else if (rmode == 2) v += R[(size_t)(m & (NPATCH - 1)) * ldc + n];
    C[(size_t)m * ldc + n] = v;
  }
}

// ---------------------------------------------------------------------------
// Head GEMM split-K partial:  part[chunk][tn][16x16] = x(16xK) * head_w^T
// One wave handles HN_GROUP adjacent N-tiles (A fragment reused 4x);
// head_w is single-use (402 MB > L2) -> non-temporal loads.
// ---------------------------------------------------------------------------
__global__ void head_partial_kernel(const float* __restrict__ x,
                                    const float* __restrict__ W,
                                    float* __restrict__ part) {
  const int ngrp = NTILES_HN / HN_GROUP;       // 48
  const int wave = blockIdx.x * (blockDim.x >> 5) + (threadIdx.x >> 5);
  if (wave >= ngrp * NCHUNKS) return;
  const int g  = wave % ngrp;
  const int ch = wave / ngrp;
  const int lane = threadIdx.x & 31;
  const int l15 = lane & 15;
  const int kq  = (lane >> 4) << 2;

  const float* Ap = x + l15 * HEAD_K + ch * KCHUNK + kq;
  const float* Wp[HN_GROUP];
#pragma unroll
  for (int j = 0; j < HN_GROUP; ++j)
    Wp[j] = W + (size_t)((g * HN_GROUP + j) * 16 + l15) * HEAD_K + ch * KCHUNK + kq;

  v8f acc[HN_GROUP] = {{}, {}, {}, {}};
  for (int k = 0; k < KCHUNK; k += 8) {
    v4f a4 = *(const v4f*)(Ap + k);
    v2f alo = vlo(a4), ahi = vhi(a4);
#pragma unroll
    for (int j = 0; j < HN_GROUP; ++j) {
      v4f b4 = __builtin_nontemporal_load((const v4f*)(Wp[j] + k));
      acc[j] = wmma_f32_4(alo, vlo(b4), acc[j]);
      acc[j] = wmma_f32_4(ahi, vhi(b4), acc[j]);
    }
  }
  const int hi = (lane >> 4) << 3;
#pragma unroll
  for (int j = 0; j < HN_GROUP; ++j) {
    float* dst = part + (size_t)(ch * NTILES_HN + g * HN_GROUP + j) * 256;
#pragma unroll
    for (int r = 0; r < 8; ++r) dst[(r + hi) * 16 + l15] = acc[j][r];
  }
}

__global__ void head_reduce_kernel(const float* __restrict__ part,
                                   const float* __restrict__ bias,
                                   float* __restrict__ out) {
  const int idx = blockIdx.x * blockDim.x + threadIdx.x;  // 16*3072
  if (idx >= BATCH * HEAD_N) return;
  const int b = idx / HEAD_N, col = idx % HEAD_N;
  const int tn = col >> 4;
  const int e = b * 16 + (col & 15);
  float acc = bias[col];
  for (int ch = 0; ch < NCHUNKS; ++ch)
    acc += part[(size_t)(ch * NTILES_HN + tn) * 256 + e];
  out[idx] = acc;   // (B, V, F_LEN) flat
}

// ---------------------------------------------------------------------------
// Elementwise / small kernels
// ---------------------------------------------------------------------------
__global__ void gather_patches_kernel(const float* __restrict__ ids,
                                      float* __restrict__ patches) {
  const int idx = blockIdx.x * blockDim.x + threadIdx.x;  // 2048*512
  if (idx >= MROWS * PATCH_K) return;
  const int col = idx & (PATCH_K - 1);
  const int m = idx >> 9;
  const int v = col >> 4, t = col & 15;
  const int p = m & (NPATCH - 1), b = m >> 7;
  patches[idx] = ids[(size_t)(b * VDIM + v) * LSEQ + p * P_LEN + t];
}

// causal depthwise conv (D_CONV=4, left pad 3) + bias + SiLU
__global__ void conv_silu_kernel(const float* __restrict__ xr,
                                 const float* __restrict__ cw,
                                 const float* __restrict__ cb,
                                 float* __restrict__ out) {
  const int idx = blockIdx.x * blockDim.x + threadIdx.x;  // 2048*512
  if (idx >= MROWS * D_INNER) return;
  const int d = idx & (D_INNER - 1);
  const int m = idx >> 9;
  const int p = m & (NPATCH - 1), b = m >> 7;
  float acc = cb[d];
#pragma unroll
  for (int j = 0; j < D_CONV; ++j) {
    const int l = p - (D_CONV - 1) + j;
    if (l >= 0)
      acc += cw[d * D_CONV + j] * xr[(size_t)((b << 7) + l) * (2 * D_INNER) + d];
  }
  out[idx] = siluf(acc);
}

__global__ void softplus_kernel(float* __restrict__ x, int n) {
  const int idx = blockIdx.x * blockDim.x + threadIdx.x;
  if (idx >= n) return;
  const float v = x[idx];
  x[idx] = (v > 20.0f) ? v : log1pf(__expf(v));
}

// selective scan: block per batch, thread per d_inner channel (512 threads)
__global__ void scan_kernel(const float* __restrict__ u,
                            const float* __restrict__ delta,
                            const float* __restrict__ xdbl,   // [m][48]
                            const float* __restrict__ A_log,
                            const float* __restrict__ Dp,
                            float* __restrict__ y) {
  const int b = blockIdx.x;
  const int d = threadIdx.x;
  __shared__ float sB[D_STATE], sC[D_STATE];
  float A[D_STATE], st[D_STATE];
#pragma unroll
  for (int n = 0; n < D_STATE; ++n) {
    A[n]  = -__expf(A_log[d * D_STATE + n]);
    st[n] = 0.0f;
  }
  const float Dd = Dp[d];
  for (int l = 0; l < NPATCH; ++l) {
    const int m = b * NPATCH + l;
    if (d < D_STATE)           sB[d]           = xdbl[m * 48 + DT_RANK + d];
    else if (d < 2 * D_STATE)  sC[d - D_STATE] = xdbl[m * 48 + DT_RANK + D_STATE + (d - D_STATE)];
    __syncthreads();
    const float dl = delta[(size_t)m * D_INNER + d];
    const float uu = u[(size_t)m * D_INNER + d];
    float acc = 0.0f;
#pragma unroll
    for (int n = 0; n < D_STATE; ++n) {
      const float dA = __expf(dl * A[n]);
      st[n] = dA * st[n] + dl * sB[n] * uu;
      acc += st[n] * sC[n];
    }
    y[(size_t)m * D_INNER + d] = acc + uu * Dd;
    __syncthreads();
  }
}

// g = y * silu(res)  (res = second half of xr rows)
__global__ void gate_kernel(const float* __restrict__ y,
                            const float* __restrict__ xr,
                            float* __restrict__ g) {
  const int idx = blockIdx.x * blockDim.x + threadIdx.x;
  if (idx >= MROWS * D_INNER) return;
  const int d = idx & (D_INNER - 1);
  const int m = idx >> 9;
  const float r = xr[(size_t)m * (2 * D_INNER) + D_INNER + d];
  g[idx] = y[idx] * siluf(r);
}

// channel attention: block per batch, 256 threads (one per channel)
__global__ void channel_attn_kernel(float* __restrict__ x,
                                    const float* __restrict__ w1,
                                    const float* __restrict__ b1,
                                    const float* __restrict__ w2,
                                    const float* __restrict__ b2) {
  const int b = blockIdx.x;
  const int n = threadIdx.x;           // 0..255
  __shared__ float s_avg[D_MODEL], s_mx[D_MODEL], s_h[64];
  float* xb = x + (size_t)b * NPATCH * D_MODEL;
  float acc = 0.0f, mx = -3.4e38f;
  for (int p = 0; p < NPATCH; ++p) {
    const float v = xb[p * D_MODEL + n];
    acc += v;
    mx = fmaxf(mx, v);
  }
  s_avg[n] = acc * (1.0f / NPATCH);
  s_mx[n]  = mx;
  __syncthreads();
  if (n < 64) {   // 32 hidden units x {avg, max}
    const int j = n & 31;
    const float* src = (n >= 32) ? s_mx : s_avg;
    float h = b1[j];
    for (int c = 0; c < D_MODEL; ++c) h += w1[j * D_MODEL + c] * src[c];
    s_h[n] = fmaxf(h, 0.0f);
  }
  __syncthreads();
  float f = 2.0f * b2[n];              // bias appears in both branches
  for (int j = 0; j < 32; ++j) f += w2[n * 32 + j] * (s_h[j] + s_h[32 + j]);
  const float attn = 1.0f / (1.0f + __expf(-f));
  for (int p = 0; p < NPATCH; ++p) xb[p * D_MODEL + n] *= attn;
}

// RMSNorm: block per token row, 256 threads
__global__ void rmsnorm_kernel(float* __restrict__ x,
                               const float* __restrict__ w) {
  const int row = blockIdx.x;
  const int n = threadIdx.x;
  __shared__ float red[D_MODEL];
  const float v = x[(size_t)row * D_MODEL + n];
  red[n] = v * v;
  __syncthreads();
  for (int s = D_MODEL / 2; s > 0; s >>= 1) {
    if (n < s) red[n] += red[n + s];
    __syncthreads();
  }
  const float scale = rsqrtf(red[0] * (1.0f / D_MODEL) + EPSF);
  x[(size_t)row * D_MODEL + n] = v * scale * w[n];
}

// ---------------------------------------------------------------------------
// Host orchestration
// ---------------------------------------------------------------------------
static inline void launch_gemm(const float* A, int lda, const float* W, int ldw,
                               const float* bias, const float* R, int rmode,
                               float* C, int ldc, int M, int N, int K,
                               hipStream_t s) {
  const int tiles = (M >> 4) * (N >> 4);
  const int blocks = (tiles + 7) >> 3;   // 8 waves / 256-thread block
  gemm_wmma_kernel<<<blocks, 256, 0, s>>>(A, lda, W, ldw, bias, R, rmode,
                                          C, ldc, M, N, K);
}

extern "C" void kernel_launch(void* const* d_in, const int* in_sizes, int n_in,
                              void* d_out, int out_size, void* d_ws, size_t ws_size,
                              hipStream_t stream) {
  const float* input_ids = (const float*)d_in[0];
  const float* pos_enc   = (const float*)d_in[1];
  const float* patch_w   = (const float*)d_in[2];
  const float* patch_b   = (const float*)d_in[3];
  const float* in_w      = (const float*)d_in[4];
  const float* conv_w    = (const float*)d_in[5];
  const float* conv_b    = (const float*)d_in[6];
  const float* xproj_w   = (const float*)d_in[7];
  const float* dt_w      = (const float*)d_in[8];
  const float* dt_b      = (const float*)d_in[9];
  const float* A_log     = (const float*)d_in[10];
  const float* D_param   = (const float*)d_in[11];
  const float* out_w     = (const float*)d_in[12];
  const float* ca_w1     = (const float*)d_in[13];
  const float* ca_b1     = (const float*)d_in[14];
  const float* ca_w2     = (const float*)d_in[15];
  const float* ca_b2     = (const float*)d_in[16];
  const float* norm_w    = (const float*)d_in[17];
  const float* normf_w   = (const float*)d_in[18];
  const float* head_w    = (const float*)d_in[19];
  const float* head_b    = (const float*)d_in[20];
  float* out = (float*)d_out;
  float* ws  = (float*)d_ws;

  // workspace layout (floats)
  float* x      = ws;                                    // 2048*256
  float* patch  = x      + (size_t)MROWS * D_MODEL;      // 2048*512
  float* xr     = patch  + (size_t)MROWS * PATCH_K;      // 2048*1024
  float* xxc    = xr     + (size_t)MROWS * 2 * D_INNER;  // 2048*512
  float* xdbl   = xxc    + (size_t)MROWS * D_INNER;      // 2048*48
  float* delta  = xdbl   + (size_t)MROWS * 48;           // 2048*512
  float* ybuf   = delta  + (size_t)MROWS * D_INNER;      // 2048*512
  float* gbuf   = ybuf   + (size_t)MROWS * D_INNER;      // 2048*512
  float* part   = patch;  // head partials reuse dead mamba scratch (24 MB)

  const int EW = MROWS * D_INNER;           // 1M elementwise threads
  const int EWB = (EW + 255) / 256;

  // 1. patch extraction + embedding (+pos encoding fused)
  gather_patches_kernel<<<EWB, 256, 0, stream>>>(input_ids, patch);
  launch_gemm(patch, PATCH_K, patch_w, PATCH_K, patch_b, pos_enc, /*pos*/2,
              x, D_MODEL, MROWS, D_MODEL, PATCH_K, stream);

  // 2. mamba stack
  for (int k = 0; k < NB; ++k) {
    const float* inw  = in_w   + (size_t)k * 2 * D_INNER * D_MODEL;
    const float* cw   = conv_w + (size_t)k * D_INNER * D_CONV;
    const float* cb   = conv_b + (size_t)k * D_INNER;
    const float* xpw  = xproj_w+ (size_t)k * 48 * D_INNER;
    const float* dtw  = dt_w   + (size_t)k * D_INNER * DT_RANK;
    const float* dtb  = dt_b   + (size_t)k * D_INNER;
    const float* Al   = A_log  + (size_t)k * D_INNER * D_STATE;
    const float* Dpar = D_param+ (size_t)k * D_INNER;
    const float* ow   = out_w  + (size_t)k * D_MODEL * D_INNER;

    launch_gemm(x, D_MODEL, inw, D_MODEL, nullptr, nullptr, 0,
                xr, 2 * D_INNER, MROWS, 2 * D_INNER, D_MODEL, stream);
    conv_silu_kernel<<<EWB, 256, 0, stream>>>(xr, cw, cb, xxc);
    launch_gemm(xxc, D_INNER, xpw, D_INNER, nullptr, nullptr, 0,
                xdbl, 48, MROWS, 48, D_INNER, stream);
    launch_gemm(xdbl, 48, dtw, DT_RANK, dtb, nullptr, 0,
                delta, D_INNER, MROWS, D_INNER, DT_RANK, stream);
    softplus_kernel<<<EWB, 256, 0, stream>>>(delta, EW);
    scan_kernel<<<BATCH, D_INNER, 0, stream>>>(xxc, delta, xdbl, Al, Dpar, ybuf);
    gate_kernel<<<EWB, 256, 0, stream>>>(ybuf, xr, gbuf);
    launch_gemm(gbuf, D_INNER, ow, D_INNER, nullptr, x, /*full resid*/1,
                x, D_MODEL, MROWS, D_MODEL, D_INNER, stream);

    if (k & 1) {  // after each pair of blocks: channel attention + rmsnorm
      const int i = k >> 1;
      channel_attn_kernel<<<BATCH, D_MODEL, 0, stream>>>(
          x, ca_w1 + (size_t)i * 32 * D_MODEL, ca_b1 + (size_t)i * 32,
          ca_w2 + (size_t)i * D_MODEL * 32, ca_b2 + (size_t)i * D_MODEL);
      rmsnorm_kernel<<<MROWS, D_MODEL, 0, stream>>>(x, norm_w + (size_t)i * D_MODEL);
    }
  }

  // 3. final norm + head (split-K over the 402 MB weight, deterministic reduce)
  rmsnorm_kernel<<<MROWS, D_MODEL, 0, stream>>>(x, normf_w);
  {
    const int waves = (NTILES_HN / HN_GROUP) * NCHUNKS;   // 6144
    head_partial_kernel<<<(waves + 7) / 8, 256, 0, stream>>>(x, head_w, part);
    const int n = BATCH * HEAD_N;
    head_reduce_kernel<<<(n + 255) / 256, 256, 0, stream>>>(part, head_b, out);
  }
}